// MLPPredictor_21303037788665
// MI455X (gfx1250) — compile-verified
//
#include <hip/hip_runtime.h>
#include <hip/hip_bf16.h>

typedef __attribute__((ext_vector_type(2))) float v2f;
typedef __attribute__((ext_vector_type(8))) float v8f;

#define KDIM 64    // DU == DI == 64
#define HH   128   // hidden size
#define TU   16    // users per score-tile
#define TI   64    // items per score-tile
#define PAD  1     // LDS row padding (floats) to break bank conflicts

// ---------------------------------------------------------------------------
// Stage 1/2: Y[M,N] = X[M,K] @ W[N,K]^T + bias[N]   (K = KDIM = 64, fp32 WMMA)
// One wave (32 threads) owns one 16x16 tile of Y, chaining 16 k=4 WMMA steps.
//
// A (16x4 f32): lane l -> row m = l%16; VGPR j holds K = 2*(l/16) + j
// B (4x16 f32): lane l -> col n = l%16; VGPR j holds K = 2*(l/16) + j
//               B[k][n] = W^T[k][n] = W[n][k]  -> contiguous pair in W's row
// C/D (16x16 f32): VGPR r -> m = r + 8*(l/16), n = l%16
// ---------------------------------------------------------------------------
__global__ __launch_bounds__(32) void proj_wmma_kernel(
    const float* __restrict__ X, const float* __restrict__ W,
    const float* __restrict__ bias, float* __restrict__ Y, int N)
{
  const int lane = threadIdx.x;        // 0..31, full wave32
  const int MB   = blockIdx.x * 16;
  const int NB   = blockIdx.y * 16;
  const int half = lane >> 4;          // 0 | 1
  const int lid  = lane & 15;

  const float* xrow = X + (MB + lid) * KDIM + half * 2;  // A fetch base
  const float* wrow = W + (NB + lid) * KDIM + half * 2;  // B fetch base

  v8f c = {0.f, 0.f, 0.f, 0.f, 0.f, 0.f, 0.f, 0.f};
#pragma unroll
  for (int k0 = 0; k0 < KDIM; k0 += 4) {
    v2f a = *(const v2f*)(xrow + k0);   // 8-byte aligned pair (K even offsets)
    v2f b = *(const v2f*)(wrow + k0);
    // D = A x B + C  (fp32, round-to-nearest-even)
    c = __builtin_amdgcn_wmma_f32_16x16x4_f32(
        /*neg_a=*/false, a, /*neg_b=*/false, b,
        /*c_mod=*/(short)0, c, /*reuse_a=*/false, /*reuse_b=*/false);
  }

  const float bn = bias[NB + lid];     // n = lid for every row this lane holds
#pragma unroll
  for (int r = 0; r < 8; ++r) {
    const int m = r + 8 * half;
    Y[(MB + m) * N + (NB + lid)] = c[r] + bn;
  }
}

// ---------------------------------------------------------------------------
// Stage 3: scores[u,i] = sum_h relu(uo[u,h] + io[i,h]) * w2[h] + b2
// Block = 256 threads handles a TU x TI tile. Tiles staged in LDS with +1
// float row padding (row reads across lanes stride 129 -> conflict-free).
// Each thread computes 4 outputs (same item, users ug, ug+4, ug+8, ug+12),
// so si-row and w2 reads amortize 4x and su reads are wave-broadcasts.
// ---------------------------------------------------------------------------
__global__ __launch_bounds__(256) void score_relu_dot_kernel(
    const float* __restrict__ uo, const float* __restrict__ io,
    const float* __restrict__ w2, const float* __restrict__ b2p,
    float* __restrict__ out, int I)
{
  __shared__ float su[TU * (HH + PAD)];
  __shared__ float si[TI * (HH + PAD)];
  __shared__ float sw[HH];

  const int t  = threadIdx.x;
  const int UB = blockIdx.x * TU;
  const int IB = blockIdx.y * TI;

  // Fill su (16x128) with b128 loads: 512 float4s, 2 per thread.
  {
    const float4* g = (const float4*)(uo + UB * HH);
    for (int f4 = t; f4 < (TU * HH) / 4; f4 += 256) {
      float4 v = g[f4];
      const int e = f4 * 4;
      float* d = su + (e >> 7) * (HH + PAD) + (e & (HH - 1));
      d[0] = v.x; d[1] = v.y; d[2] = v.z; d[3] = v.w;
    }
  }
  // Fill si (64x128): 2048 float4s, 8 per thread.
  {
    const float4* g = (const float4*)(io + IB * HH);
    for (int f4 = t; f4 < (TI * HH) / 4; f4 += 256) {
      float4 v = g[f4];
      const int e = f4 * 4;
      float* d = si + (e >> 7) * (HH + PAD) + (e & (HH - 1));
      d[0] = v.x; d[1] = v.y; d[2] = v.z; d[3] = v.w;
    }
  }
  if (t < HH) sw[t] = w2[t];
  __syncthreads();

  const int il = t & (TI - 1);   // item lane within tile
  const int ug = t >> 6;         // user group 0..3

  const float* srow = si + il * (HH + PAD);
  const float* u0r  = su + (ug +  0) * (HH + PAD);
  const float* u1r  = su + (ug +  4) * (HH + PAD);
  const float* u2r  = su + (ug +  8) * (HH + PAD);
  const float* u3r  = su + (ug + 12) * (HH + PAD);

  float a0 = 0.f, a1 = 0.f, a2 = 0.f, a3 = 0.f;
#pragma unroll 8
  for (int h = 0; h < HH; ++h) {
    const float b = srow[h];     // conflict-free (pad), distinct row per lane
    const float w = sw[h];       // broadcast
    a0 = fmaf(fmaxf(u0r[h] + b, 0.f), w, a0);
    a1 = fmaf(fmaxf(u1r[h] + b, 0.f), w, a1);
    a2 = fmaf(fmaxf(u2r[h] + b, 0.f), w, a2);
    a3 = fmaf(fmaxf(u3r[h] + b, 0.f), w, a3);
  }

  const float bb = b2p[0];
  float* o = out + (UB + ug) * I + (IB + il);
  o[0 * 4 * I] = a0 + bb;
  o[1 * 4 * I] = a1 + bb;
  o[2 * 4 * I] = a2 + bb;
  o[3 * 4 * I] = a3 + bb;
}

// ---------------------------------------------------------------------------
// inputs: users(U,64) items(I,64) Wu(128,64) bu(128) Wi(128,64) bi(128)
//         w2(128) b2()
// ---------------------------------------------------------------------------
extern "C" void kernel_launch(void* const* d_in, const int* in_sizes, int n_in,
                              void* d_out, int out_size, void* d_ws, size_t ws_size,
                              hipStream_t stream) {
  const float* users = (const float*)d_in[0];
  const float* items = (const float*)d_in[1];
  const float* Wu    = (const float*)d_in[2];
  const float* bu    = (const float*)d_in[3];
  const float* Wi    = (const float*)d_in[4];
  const float* bi    = (const float*)d_in[5];
  const float* w2    = (const float*)d_in[6];
  const float* b2    = (const float*)d_in[7];
  float* out = (float*)d_out;

  const int U = 1024, I = 2048;

  float* uo = (float*)d_ws;      // U*HH fp32  (512 KB)
  float* io = uo + U * HH;       // I*HH fp32  (1 MB)

  dim3 b32(32);
  dim3 g_u(U / 16, HH / 16);     // 64 x 8 tiles
  proj_wmma_kernel<<<g_u, b32, 0, stream>>>(users, Wu, bu, uo, HH);
  dim3 g_i(I / 16, HH / 16);     // 128 x 8 tiles
  proj_wmma_kernel<<<g_i, b32, 0, stream>>>(items, Wi, bi, io, HH);

  dim3 b256(256);
  dim3 g_s(U / TU, I / TI);      // 64 x 32 blocks
  score_relu_dot_kernel<<<g_s, b256, 0, stream>>>(uo, io, w2, b2, out, I);
}